// KANLayer_81690277970277
// MI455X (gfx1250) — compile-verified
//
#include <hip/hip_runtime.h>

typedef __attribute__((ext_vector_type(2))) float v2f;
typedef __attribute__((ext_vector_type(8))) float v8f;

#define B_SZ    4096
#define N_IN    512
#define N_BASIS 8
#define N_SLICE 16
// 1/sqrt(2*pi)
#define INV_KN  0.3989422804014327f

// One wave per 16-row tile of the flattened (B*N_IN, N_SLICE) output.
// D(16x16) = A(16x8 basis values) x B(8x16 Wbb), done as two
// V_WMMA_F32_16X16X4_F32 ops (K=0..3, K=4..7), then fused SiLU/Wc/Wb combine.
__global__ __launch_bounds__(256) void kan_wmma_kernel(
    const float* __restrict__ x,      // (B_SZ, N_IN)
    const float* __restrict__ Wbb,    // (N_BASIS, N_SLICE)
    const float* __restrict__ Wb,     // (N_IN, N_SLICE)
    const float* __restrict__ Wc,     // (N_IN)
    const float* __restrict__ q,      // (N_IN, N_BASIS)
    const float* __restrict__ mprm,   // (N_BASIS)
    float* __restrict__ out)          // (B_SZ*N_IN, N_SLICE)
{
    const int lane    = threadIdx.x & 31;
    const int wave    = threadIdx.x >> 5;
    const int tile    = blockIdx.x * 8 + wave;
    const int rowbase = tile * 16;              // flat row (b*N_IN + n) base
    const int mrow    = lane & 15;              // A-matrix row this lane feeds
    const int half    = lane >> 4;              // lane group selects K pairs
    const int ncol    = lane & 15;              // B/D column this lane holds
    const int kb      = 2 * half;               // first K this lane holds

    const int row   = rowbase + mrow;
    const int n_idx = row & (N_IN - 1);         // N_IN=512 pow2; 16 | 512 so no wrap in-tile

    // ---- B matrix (Wbb) in WMMA layout: lane = N col, VGPR j = K row (2*half + j)
    v2f b0, b1;
    b0.x = Wbb[(kb + 0) * N_SLICE + ncol];
    b0.y = Wbb[(kb + 1) * N_SLICE + ncol];
    b1.x = Wbb[(kb + 4) * N_SLICE + ncol];
    b1.y = Wbb[(kb + 5) * N_SLICE + ncol];

    // ---- A matrix: Gaussian basis phsi(row, k) for this lane's 4 K slots
    const float xv = x[row];
    v2f a0, a1;
    {
        const float q0 = q[n_idx * N_BASIS + kb + 0];
        const float q1 = q[n_idx * N_BASIS + kb + 1];
        const float q2 = q[n_idx * N_BASIS + kb + 4];
        const float q3 = q[n_idx * N_BASIS + kb + 5];
        const float m0 = mprm[kb + 0];
        const float m1 = mprm[kb + 1];
        const float m2 = mprm[kb + 4];
        const float m3 = mprm[kb + 5];
        const float r0 = __builtin_amdgcn_rcpf(q0);
        const float r1 = __builtin_amdgcn_rcpf(q1);
        const float r2 = __builtin_amdgcn_rcpf(q2);
        const float r3 = __builtin_amdgcn_rcpf(q3);
        float z;
        z = (xv - m0) * r0;  a0.x = __expf(-0.5f * z * z) * r0 * INV_KN;
        z = (xv - m1) * r1;  a0.y = __expf(-0.5f * z * z) * r1 * INV_KN;
        z = (xv - m2) * r2;  a1.x = __expf(-0.5f * z * z) * r2 * INV_KN;
        z = (xv - m3) * r3;  a1.y = __expf(-0.5f * z * z) * r3 * INV_KN;
    }

    // ---- Bmat(16x16) = phsi x Wbb via two chained WMMAs (K=0..3 then 4..7)
    v8f acc = {};
    acc = __builtin_amdgcn_wmma_f32_16x16x4_f32(
        /*neg_a=*/false, a0, /*neg_b=*/false, b0,
        /*c_mod=*/(short)0, acc, /*reuse_a=*/false, /*reuse_b=*/false);
    acc = __builtin_amdgcn_wmma_f32_16x16x4_f32(
        false, a1, false, b1, (short)0, acc, false, false);

    // ---- SiLU branch for this lane's own row: silu(x) * Wc[n]
    // v_rcp_f32 instead of the full IEEE divide expansion.
    const float sig = __builtin_amdgcn_rcpf(1.0f + __expf(-xv));
    const float sil = xv * sig * Wc[n_idx];

    // ---- Combine + store. D VGPR r holds (row = r + 8*half, col = ncol).
    const int nbase = rowbase & (N_IN - 1);
#pragma unroll
    for (int r = 0; r < 8; ++r) {
        const int   rr   = r + 8 * half;                 // tile-local output row
        const float s_rr = __shfl(sil, rr, 32);          // row scalar from lane rr
        const int   n_rr = nbase + rr;
        const float wb   = Wb[n_rr * N_SLICE + ncol];
        out[(size_t)(rowbase + rr) * N_SLICE + ncol] = s_rr + acc[r] * wb;
    }
}

extern "C" void kernel_launch(void* const* d_in, const int* in_sizes, int n_in,
                              void* d_out, int out_size, void* d_ws, size_t ws_size,
                              hipStream_t stream) {
    (void)in_sizes; (void)n_in; (void)out_size; (void)d_ws; (void)ws_size;
    const float* x   = (const float*)d_in[0];
    const float* Wbb = (const float*)d_in[1];
    const float* Wb  = (const float*)d_in[2];
    const float* Wc  = (const float*)d_in[3];
    const float* q   = (const float*)d_in[4];
    const float* m   = (const float*)d_in[5];
    float* out = (float*)d_out;

    const int tiles  = (B_SZ * N_IN) / 16;   // 131072 16-row tiles
    const int blocks = tiles / 8;            // 8 waves (tiles) per 256-thread block
    kan_wmma_kernel<<<dim3(blocks), dim3(256), 0, stream>>>(x, Wbb, Wb, Wc, q, m, out);
}